// QGCost_weak_fourdvar_60945585930617
// MI455X (gfx1250) — compile-verified
//
#include <hip/hip_runtime.h>
#include <hip/hip_bf16.h>
#include <math.h>

// ---------------------------------------------------------------------------
// QG model, MI455X (gfx1250, wave32). Heavy math = 4x (512^3) f32 GEMMs/step
// implemented with V_WMMA_F32_16X16X4_F32. All state is L2-resident (~11 MB).
// ---------------------------------------------------------------------------

#define N      512
#define N2     (N * N)
#define NIN    510
#define GCONST 9.81f
#define DTSTEP 3600.0f

typedef float v2f __attribute__((ext_vector_type(2)));
typedef float v8f __attribute__((ext_vector_type(8)));

// ---------------------------------------------------------------------------
// init: h = nan_to_num(h0[0,0]); hb = h; Xp = 0; S (padded DST-I matrix);
// helmPad (padded with 1.0 to make the spectral divide a no-op in padding)
// ---------------------------------------------------------------------------
__global__ void k_init(const float* __restrict__ h0,
                       const float* __restrict__ helm,
                       float* __restrict__ h_a, float* __restrict__ hb,
                       float* __restrict__ Xp, float* __restrict__ S,
                       float* __restrict__ helmPad) {
  int idx = blockIdx.x * blockDim.x + threadIdx.x;
  if (idx >= N2) return;
  int i = idx >> 9, j = idx & (N - 1);

  float hv = h0[idx];                 // h0[0,0,:,:] is the first 512*512 slice
  hv = (hv == hv) ? hv : 0.0f;        // nan_to_num
  h_a[idx] = hv;
  hb[idx]  = hv;
  Xp[idx]  = 0.0f;

  bool in = (i < NIN) && (j < NIN);
  helmPad[idx] = in ? helm[i * NIN + j] : 1.0f;

  float sv = 0.0f;
  if (in) {
    // S[i][j] = sqrt(2/511) * sin(pi*(i+1)*(j+1)/511), exact mod-1022 reduction
    long long p = (long long)(i + 1) * (long long)(j + 1);
    int r = (int)(p % 1022);
    double ang = M_PI * (double)r / 511.0;
    sv = (float)(sqrt(2.0 / 511.0) * sin(ang));
  }
  S[idx] = sv;
}

// ---------------------------------------------------------------------------
// q0 = h2pv(h, h, f, c): qb and initial q (identical since hb == h)
// ind12 mask recomputed analytically: i<2 | j<2 | i>=509 | j>=509
// ---------------------------------------------------------------------------
__global__ void k_q0(const float* __restrict__ h, const float* __restrict__ f,
                     const float* __restrict__ c,
                     float* __restrict__ qb, float* __restrict__ qa,
                     float invdx2) {
  int idx = blockIdx.x * blockDim.x + threadIdx.x;
  if (idx >= N2) return;
  int i = idx >> 9, j = idx & (N - 1);
  float fv = f[idx], cv = c[idx], hv = h[idx];
  float gfc2 = GCONST * fv / (cv * cv);
  bool ind12 = (i < 2) || (j < 2) || (i >= NIN - 1) || (j >= NIN - 1);
  float q;
  if (ind12) {
    q = -gfc2 * hv;
  } else {
    float lap = (h[(i + 1) * N + j] + h[(i - 1) * N + j] - 2.0f * hv) * invdx2 +
                (h[i * N + j + 1] + h[i * N + j - 1] - 2.0f * hv) * invdx2;
    q = (GCONST / fv) * lap - gfc2 * hv;
    if (!(q == q)) q = 0.0f;
  }
  qb[idx] = q;
  qa[idx] = q;
}

// ---------------------------------------------------------------------------
// fused h2uv + rhs + Euler update + qin extraction
// ---------------------------------------------------------------------------
__device__ __forceinline__ float u_at(const float* __restrict__ h,
                                      const float* __restrict__ f,
                                      int i, int j, float inv4dx) {
  // valid for i in [1,511), j in [1,512)
  return -(GCONST / f[i * N + j]) *
         (h[(i + 1) * N + (j - 1)] + h[(i + 1) * N + j] -
          h[(i - 1) * N + j] - h[(i - 1) * N + (j - 1)]) * inv4dx;
}
__device__ __forceinline__ float v_at(const float* __restrict__ h,
                                      const float* __restrict__ f,
                                      int i, int j, float inv4dx) {
  // valid for i in [1,512), j in [1,511)
  return (GCONST / f[i * N + j]) *
         (h[i * N + (j + 1)] + h[(i - 1) * N + (j + 1)] -
          h[(i - 1) * N + (j - 1)] - h[i * N + (j - 1)]) * inv4dx;
}

__global__ void k_step(const float* __restrict__ h, const float* __restrict__ q1,
                       const float* __restrict__ qb, const float* __restrict__ f,
                       float* __restrict__ q2, float* __restrict__ Xp,
                       float inv4dx, float c1, float inv2dx) {
  int idx = blockIdx.x * blockDim.x + threadIdx.x;
  if (idx >= N2) return;
  int i = idx >> 9, j = idx & (N - 1);

  float r = 0.0f;
  bool ind12 = (i < 2) || (j < 2) || (i >= NIN - 1) || (j >= NIN - 1);
  if (!ind12) {  // implies 2 <= i,j <= 508, all stencils in range
    float u00 = u_at(h, f, i, j, inv4dx);
    float u01 = u_at(h, f, i, j + 1, inv4dx);
    float v00 = v_at(h, f, i, j, inv4dx);
    float v10 = v_at(h, f, i + 1, j, inv4dx);
    float uT = 0.5f * (u00 + u01);
    float vT = 0.5f * (v00 + v10);
    float up = fmaxf(uT, 0.0f), um = fminf(uT, 0.0f);
    float vp = fmaxf(vT, 0.0f), vm = fminf(vT, 0.0f);
#define Q(a, b) q1[(a) * N + (b)]
    float adv =
        -up * c1 * (2.0f * Q(i, j + 1) + 3.0f * Q(i, j) - 6.0f * Q(i, j - 1) + Q(i, j - 2)) +
         um * c1 * (Q(i, j + 2) - 6.0f * Q(i, j + 1) + 3.0f * Q(i, j) + 2.0f * Q(i, j - 1)) -
         vp * c1 * (2.0f * Q(i + 1, j) + 3.0f * Q(i, j) - 6.0f * Q(i - 1, j) + Q(i - 2, j)) +
         vm * c1 * (Q(i + 2, j) - 6.0f * Q(i + 1, j) + 3.0f * Q(i, j) + 2.0f * Q(i - 1, j));
#undef Q
    float beta = (f[(i + 1) * N + j] - f[(i - 1) * N + j]) * inv2dx;
    r = adv - beta * 0.5f * (v00 + v10);
    if (!(r == r)) r = 0.0f;  // nan_to_num
  }
  float q2v = q1[idx] + DTSTEP * r;
  q2[idx] = q2v;
  if (i >= 1 && i < N - 1 && j >= 1 && j < N - 1)
    Xp[(i - 1) * N + (j - 1)] = q2v - qb[idx];  // padding rows/cols stay zero
}

// ---------------------------------------------------------------------------
// 512x512x512 f32 GEMM: C = A*B  (optionally C = (A*B) ./ divisor)
// Block = 256 threads (8 waves), 64x64 C tile, K-tile 32, WMMA 16x16x4 f32.
// Wave w: tile-row (w&3)*16, col group (w>>2)*32, two 16x16 accumulators.
// ---------------------------------------------------------------------------
#define BM 64
#define BN 64
#define BK 32

__global__ __launch_bounds__(256) void k_gemm(const float* __restrict__ A,
                                              const float* __restrict__ B,
                                              float* __restrict__ C,
                                              const float* __restrict__ divisor) {
  __shared__ float As[BM][BK + 1];  // +1 pad: conflict-free column reads
  __shared__ float Bs[BK][BN];

  const int tid  = threadIdx.x;
  const int wave = tid >> 5;
  const int lane = tid & 31;
  const int row0 = blockIdx.x * BM;
  const int col0 = blockIdx.y * BN;

  const int tr = (wave & 3) * 16;        // wave's 16-row strip within tile
  const int cg = (wave >> 2) * 32;       // wave's 32-col group within tile
  const int m  = lane & 15;              // A/B/C lane index (M or N)
  const int kh = (lane >> 4) * 2;        // K sub-offset for upper half-wave
  const int hi = lane >> 4;              // C row half select

  v8f acc0 = {}; v8f acc1 = {};

  for (int k0 = 0; k0 < N; k0 += BK) {
    // A tile: 64x32, 8 consecutive floats per thread (2x float4)
    {
      int r = tid >> 2, c = (tid & 3) * 8;
      const float4* g = (const float4*)(A + (row0 + r) * N + k0 + c);
      float4 p0 = g[0], p1 = g[1];
      As[r][c + 0] = p0.x; As[r][c + 1] = p0.y; As[r][c + 2] = p0.z; As[r][c + 3] = p0.w;
      As[r][c + 4] = p1.x; As[r][c + 5] = p1.y; As[r][c + 6] = p1.z; As[r][c + 7] = p1.w;
    }
    // B tile: 32x64
    {
      int r = tid >> 3, c = (tid & 7) * 8;
      const float4* g = (const float4*)(B + (k0 + r) * N + col0 + c);
      float4 p0 = g[0], p1 = g[1];
      Bs[r][c + 0] = p0.x; Bs[r][c + 1] = p0.y; Bs[r][c + 2] = p0.z; Bs[r][c + 3] = p0.w;
      Bs[r][c + 4] = p1.x; Bs[r][c + 5] = p1.y; Bs[r][c + 6] = p1.z; Bs[r][c + 7] = p1.w;
    }
    __syncthreads();

#pragma unroll
    for (int kk = 0; kk < BK; kk += 4) {
      // A frag (16x4): lanes 0-15 K={kk,kk+1}, lanes 16-31 K={kk+2,kk+3}
      v2f a;
      a.x = As[tr + m][kk + kh];
      a.y = As[tr + m][kk + kh + 1];
      // B frags (4x16): VGPR0 K=kh, VGPR1 K=kh+1 per half-wave
      v2f b0, b1;
      b0.x = Bs[kk + kh][cg + m];      b0.y = Bs[kk + kh + 1][cg + m];
      b1.x = Bs[kk + kh][cg + 16 + m]; b1.y = Bs[kk + kh + 1][cg + 16 + m];
      acc0 = __builtin_amdgcn_wmma_f32_16x16x4_f32(false, a, false, b0,
                                                   (short)0, acc0, false, false);
      acc1 = __builtin_amdgcn_wmma_f32_16x16x4_f32(false, a, false, b1,
                                                   (short)0, acc1, false, false);
    }
    __syncthreads();
  }

  // C/D layout: VGPR e -> M = e (lanes 0-15) / e+8 (lanes 16-31)
  const int colA = col0 + cg + m;
  const int colB = colA + 16;
#pragma unroll
  for (int e = 0; e < 8; ++e) {
    int r = row0 + tr + hi * 8 + e;
    float vA = acc0[e], vB = acc1[e];
    if (divisor) {  // uniform branch (spectral Helmholtz divide)
      vA /= divisor[r * N + colA];
      vB /= divisor[r * N + colB];
    }
    C[r * N + colA] = vA;
    C[r * N + colB] = vB;
  }
}

// ---------------------------------------------------------------------------
// h2 = hb + pad(Inv); optional snapshot into d_out
// ---------------------------------------------------------------------------
__global__ void k_combine(const float* __restrict__ hb, const float* __restrict__ Inv,
                          float* __restrict__ h2, float* __restrict__ snap) {
  int idx = blockIdx.x * blockDim.x + threadIdx.x;
  if (idx >= N2) return;
  int i = idx >> 9, j = idx & (N - 1);
  float v = hb[idx];
  if (i >= 1 && i < N - 1 && j >= 1 && j < N - 1)
    v += Inv[(i - 1) * N + (j - 1)];
  h2[idx] = v;
  if (snap) snap[idx] = v;
}

// ---------------------------------------------------------------------------
extern "C" void kernel_launch(void* const* d_in, const int* in_sizes, int n_in,
                              void* d_out, int out_size, void* d_ws, size_t ws_size,
                              hipStream_t stream) {
  const float* h0   = (const float*)d_in[0];  // (1,15,512,512) -> use [0,0]
  const float* f    = (const float*)d_in[1];  // (512,512)
  const float* c    = (const float*)d_in[2];  // (512,512)
  const float* helm = (const float*)d_in[3];  // (510,510)
  // d_in[4] (ind12) unused: mask recomputed analytically.

  // DX on host in double, replicating _grid_constants():
  // dx == 0.05*111000 == 5550 everywhere; dy = 5550*cos(lat_j) with edge copy.
  double s = 0.0;
  for (int j = 0; j < N; ++j) {
    int jj = (j == 0) ? 1 : (j == N - 1) ? N - 2 : j;
    double lat = 30.0 + 0.05 * (double)jj;
    s += cos(lat * M_PI / 180.0);
  }
  const double DX = (5550.0 + 5550.0 * (s / (double)N)) * 0.5;
  const float inv4dx = (float)(1.0 / (4.0 * DX));
  const float c1     = (float)(1.0 / (6.0 * DX));
  const float inv2dx = (float)(1.0 / (2.0 * DX));
  const float invdx2 = (float)(1.0 / (DX * DX));

  // Workspace: 11 x 1MB buffers
  float* w = (float*)d_ws;
  float* h_a = w + 0 * N2;  float* h_b = w + 1 * N2;
  float* q_a = w + 2 * N2;  float* q_b = w + 3 * N2;
  float* hb  = w + 4 * N2;  float* qb  = w + 5 * N2;
  float* Xp  = w + 6 * N2;  // also reused as T3
  float* T1  = w + 7 * N2;  // also reused as Inv
  float* T2  = w + 8 * N2;
  float* S   = w + 9 * N2;
  float* hlm = w + 10 * N2;

  float* out = (float*)d_out;

  const int EB = 256, EG = (N2 + EB - 1) / EB;
  dim3 gGrid(N / BM, N / BN);

  k_init<<<EG, EB, 0, stream>>>(h0, helm, h_a, hb, Xp, S, hlm);
  k_q0<<<EG, EB, 0, stream>>>(h_a, f, c, qb, q_a, invdx2);

  float* h_cur = h_a; float* h_nxt = h_b;
  float* q_cur = q_a; float* q_nxt = q_b;

  // Only 25 steps are needed: output slices are hs[0] and hs[24].
  for (int sstep = 0; sstep < 25; ++sstep) {
    k_step<<<EG, EB, 0, stream>>>(h_cur, q_cur, qb, f, q_nxt, Xp,
                                  inv4dx, c1, inv2dx);
    // Inv = S * ((S*Xp*S) ./ helm) * S
    k_gemm<<<gGrid, EB, 0, stream>>>(S,  Xp, T1, nullptr);  // T1 = S*qin
    k_gemm<<<gGrid, EB, 0, stream>>>(T1, S,  T2, hlm);      // T2 = (T1*S)./helm
    k_gemm<<<gGrid, EB, 0, stream>>>(S,  T2, Xp, nullptr);  // T3 = S*T2 (reuse Xp)
    k_gemm<<<gGrid, EB, 0, stream>>>(Xp, S,  T1, nullptr);  // Inv = T3*S (reuse T1)

    float* snap = (sstep == 0) ? out : (sstep == 24) ? out + N2 : nullptr;
    k_combine<<<EG, EB, 0, stream>>>(hb, T1, h_nxt, snap);

    float* t;
    t = h_cur; h_cur = h_nxt; h_nxt = t;
    t = q_cur; q_cur = q_nxt; q_nxt = t;
  }
}